// ROI_Pooling_33200097198627
// MI455X (gfx1250) — compile-verified
//
#include <hip/hip_runtime.h>
#include <hip/hip_bf16.h>
#include <math.h>
#include <stdint.h>

#define POOLSZ 7
#define HH 50
#define WW 50
#define CC 512
#define NROI 300
#define NPIX (HH * WW)          // 2500
#define C4 (CC / 4)             // 128 float4 per bin row
#define OUT4_PER_ROI (POOLSZ * POOLSZ * C4)  // 6272 float4 per ROI

// gfx1250 async global->LDS copy path, guarded so host pass / older toolchains
// fall back to plain loads (compile must never break).
#if defined(__HIP_DEVICE_COMPILE__) && defined(__gfx1250__) && \
    __has_builtin(__builtin_amdgcn_global_load_async_to_lds_b32) && \
    __has_builtin(__builtin_amdgcn_s_wait_asynccnt)
#define USE_ASYNC_LDS 1
typedef __attribute__((address_space(1))) int* gas_i32_ptr;
typedef __attribute__((address_space(3))) int* las_i32_ptr;
#else
#define USE_ASYNC_LDS 0
#endif

// ---------------------------------------------------------------------------
// Kernel 1: channel-max. One wave32 per pixel: 32 lanes x 16 floats = 512 ch.
// Bandwidth: 5.12 MB read, 10 KB write.
// ---------------------------------------------------------------------------
__global__ __launch_bounds__(256) void fmax_kernel(const float* __restrict__ fm,
                                                   float* __restrict__ fmax) {
    const int wave = (blockIdx.x * blockDim.x + threadIdx.x) >> 5;
    const int lane = threadIdx.x & 31;
    if (wave >= NPIX) return;

    const float4* p = (const float4*)(fm + (size_t)wave * CC);
    float m = -INFINITY;
#pragma unroll
    for (int k = 0; k < 4; ++k) {
        float4 v = p[lane + k * 32];
        m = fmaxf(m, fmaxf(fmaxf(v.x, v.y), fmaxf(v.z, v.w)));
    }
    // wave32 butterfly reduction (gfx1250 is wave32-only)
#pragma unroll
    for (int off = 16; off > 0; off >>= 1)
        m = fmaxf(m, __shfl_xor(m, off, 32));
    if (lane == 0) fmax[wave] = m;
}

// ---------------------------------------------------------------------------
// Kernel 2: one block per ROI. Stage fmax (10 KB) in LDS via async-to-LDS,
// compute 7x7 bin maxima, then stream 100 KB of broadcast output per ROI
// with coalesced b128 stores.
// ---------------------------------------------------------------------------
__global__ __launch_bounds__(512) void roi_pool_kernel(const float* __restrict__ rois,
                                                       const float* __restrict__ fmax,
                                                       float* __restrict__ out) {
    __shared__ float sf[NPIX];                 // 10000 B
    __shared__ float pooled[POOLSZ * POOLSZ];
    __shared__ int hs[POOLSZ], he[POOLSZ], wst[POOLSZ], wen[POOLSZ];

    const int tid = threadIdx.x;
    const int roi = blockIdx.x;

    // --- stage fmax into LDS -------------------------------------------------
#if USE_ASYNC_LDS
    for (int i = tid; i < NPIX; i += 512) {
        __builtin_amdgcn_global_load_async_to_lds_b32(
            (gas_i32_ptr)(uintptr_t)(fmax + i),   // global src (AS1 int*)
            (las_i32_ptr)(sf + i),                // LDS dst (AS3 int*)
            /*offset=*/0, /*cpol=*/0);
    }
    __builtin_amdgcn_s_wait_asynccnt(0);
#else
    for (int i = tid; i < NPIX; i += 512) sf[i] = fmax[i];
#endif
    __syncthreads();

    // --- bin bounds (exactly mirrors reference integer math) -----------------
    if (tid < POOLSZ) {
        const float s = 1.0f / 16.0f;   // SUB_SAMPLE
        const int x1 = (int)(rois[roi * 5 + 1] * s);
        const int y1 = (int)(rois[roi * 5 + 2] * s);
        const int x2 = (int)(rois[roi * 5 + 3] * s);
        const int y2 = (int)(rois[roi * 5 + 4] * s);
        const int rh = y2 - y1 + 1;
        const int rw = x2 - x1 + 1;
        const int i = tid;
        int a;
        a = y1 + (i * rh) / POOLSZ;                      hs[i]  = min(max(a, 0), HH);
        a = y1 + ((i + 1) * rh + POOLSZ - 1) / POOLSZ;   he[i]  = min(max(a, 0), HH);
        a = x1 + (i * rw) / POOLSZ;                      wst[i] = min(max(a, 0), WW);
        a = x1 + ((i + 1) * rw + POOLSZ - 1) / POOLSZ;   wen[i] = min(max(a, 0), WW);
    }
    __syncthreads();

    // --- 49 bin maxima from LDS ---------------------------------------------
    if (tid < POOLSZ * POOLSZ) {
        const int bi = tid / POOLSZ;
        const int bj = tid % POOLSZ;
        float m = -INFINITY;                  // empty bin -> -inf, like reference
        const int r0 = hs[bi], r1 = he[bi];
        const int c0 = wst[bj], c1 = wen[bj];
        for (int r = r0; r < r1; ++r) {
            const float* row = sf + r * WW;
            for (int c = c0; c < c1; ++c) m = fmaxf(m, row[c]);
        }
        pooled[tid] = m;
    }
    __syncthreads();

    // --- broadcast write: 49 bins x 128 float4, fully coalesced --------------
    float4* o4 = (float4*)out + (size_t)roi * OUT4_PER_ROI;
    for (int idx = tid; idx < OUT4_PER_ROI; idx += 512) {
        const float v = pooled[idx >> 7];     // 128 float4 per bin
        o4[idx] = make_float4(v, v, v, v);
    }
}

extern "C" void kernel_launch(void* const* d_in, const int* in_sizes, int n_in,
                              void* d_out, int out_size, void* d_ws, size_t ws_size,
                              hipStream_t stream) {
    const float* rois = (const float*)d_in[0];   // (300, 5) float32
    const float* fm   = (const float*)d_in[1];   // (50, 50, 512) float32
    float* out  = (float*)d_out;                 // (300, 7, 7, 512) float32
    float* fmax = (float*)d_ws;                  // 2500 floats scratch

    const int waves  = NPIX;                     // one wave32 per pixel
    const int blocks = (waves * 32 + 255) / 256; // 313
    fmax_kernel<<<blocks, 256, 0, stream>>>(fm, fmax);
    roi_pool_kernel<<<NROI, 512, 0, stream>>>(rois, fmax, out);
}